// OuterProduct_38019050504722
// MI455X (gfx1250) — compile-verified
//
#include <hip/hip_runtime.h>

typedef __attribute__((ext_vector_type(2))) float v2f;
typedef __attribute__((ext_vector_type(8))) float v8f;

#define D_ATOM   128
#define D_HID    32
#define D_PAIR   64
#define NLOC     384
#define TWO_DHID 64

// ---------------- stage 1: AB[n][h] = op_mask * (sum_d m[n][d]*W_in[h][d] + b_in[h])
__global__ __launch_bounds__(256)
void stage1_linear_in(const float* __restrict__ m,
                      const float* __restrict__ W_in,
                      const float* __restrict__ b_in,
                      const float* __restrict__ op_mask,
                      float* __restrict__ AB) {
  int idx = blockIdx.x * blockDim.x + threadIdx.x;   // 384*64 threads
  int n = idx >> 6;
  int h = idx & 63;
  const float* mr = m + n * D_ATOM;
  const float* wr = W_in + h * D_ATOM;
  float s = 0.f;
#pragma unroll 8
  for (int d = 0; d < D_ATOM; ++d) s = fmaf(mr[d], wr[d], s);
  AB[idx] = (s + b_in[h]) * op_mask[0];
}

// ---------------- stage 2: T[j][x][p] = sum_y AB[j][32+y] * W_out[p][x*32+y]
__global__ __launch_bounds__(256)
void stage2_bw(const float* __restrict__ AB,
               const float* __restrict__ W_out,
               float* __restrict__ T) {
  int idx = blockIdx.x * blockDim.x + threadIdx.x;   // 384*32*64 threads
  int j   = idx >> 11;
  int rem = idx & 2047;
  int x   = rem >> 6;
  int p   = rem & 63;
  const float* br = AB + j * TWO_DHID + D_HID;       // b[j][*]
  const float* wr = W_out + p * (D_HID * D_HID) + x * D_HID;
  float s = 0.f;
#pragma unroll
  for (int y = 0; y < D_HID; ++y) s = fmaf(br[y], wr[y], s);
  T[idx] = s;
}

// ---------------- stage 3: z[i][j][p] = op_norm * (sum_x A[i][x]*T[j][x][p] + b_out[p])
// grid: (NLOC j-cols, 3 i-blocks of 128); block 256 = 8 waves.
// Each wave: fixed 16-row i-subtile, loops 4 p-subtiles of 16, K=32 via 8x WMMA f32 16x16x4.
__global__ __launch_bounds__(256)
void stage3_wmma(const float* __restrict__ AB,
                 const float* __restrict__ T,
                 const float* __restrict__ b_out,
                 const float* __restrict__ op_norm,
                 float* __restrict__ z) {
  const int j     = blockIdx.x;
  const int iblk  = blockIdx.y;
  const int wave  = threadIdx.x >> 5;
  const int lane  = threadIdx.x & 31;
  const int lhalf = lane >> 4;        // 0: lanes 0-15, 1: lanes 16-31
  const int l16   = lane & 15;
  const int i0    = iblk * 128 + wave * 16;
  const float* Tj = T + j * (D_HID * D_PAIR);
  const float  norm = op_norm[0];

  // Preload A fragments for all 8 K-steps.
  // ISA f32 A 16x4 layout: lanes 0-15 -> {K=k0,k0+1}, lanes 16-31 -> {K=k0+2,k0+3}; M = lane&15.
  v2f afrag[8];
  const float* ar = AB + (i0 + l16) * TWO_DHID;
#pragma unroll
  for (int ks = 0; ks < 8; ++ks) {
    int kb = ks * 4 + lhalf * 2;
    afrag[ks].x = ar[kb];
    afrag[ks].y = ar[kb + 1];
  }

#pragma unroll
  for (int pt = 0; pt < 4; ++pt) {
    const int p0 = pt * 16;
    v8f acc = {};
#pragma unroll
    for (int ks = 0; ks < 8; ++ks) {
      int kb = ks * 4 + lhalf * 2;
      // B 4x16 layout: lane = N (mod 16); lane half selects K pair (mirrors A).
      v2f bfrag;
      bfrag.x = Tj[kb * D_PAIR + p0 + l16];
      bfrag.y = Tj[(kb + 1) * D_PAIR + p0 + l16];
      acc = __builtin_amdgcn_wmma_f32_16x16x4_f32(
          /*neg_a=*/false, afrag[ks], /*neg_b=*/false, bfrag,
          /*c_mod=*/(short)0, acc, /*reuse_a=*/false, /*reuse_b=*/false);
    }
    const float bo = b_out[p0 + l16];
    // C/D layout: VGPR r -> M = r (lanes 0-15) or r+8 (lanes 16-31); N = lane&15.
#pragma unroll
    for (int r = 0; r < 8; ++r) {
      int i = i0 + r + lhalf * 8;
      z[(i * NLOC + j) * D_PAIR + p0 + l16] = (acc[r] + bo) * norm;
    }
  }
}

extern "C" void kernel_launch(void* const* d_in, const int* in_sizes, int n_in,
                              void* d_out, int out_size, void* d_ws, size_t ws_size,
                              hipStream_t stream) {
  // setup_inputs order: m, nlist, op_mask, op_norm, W_in, b_in, W_out, b_out
  const float* m       = (const float*)d_in[0];
  // d_in[1] = nlist (int) — unused by the reference computation
  const float* op_mask = (const float*)d_in[2];
  const float* op_norm = (const float*)d_in[3];
  const float* W_in    = (const float*)d_in[4];
  const float* b_in    = (const float*)d_in[5];
  const float* W_out   = (const float*)d_in[6];
  const float* b_out   = (const float*)d_in[7];
  float* z = (float*)d_out;

  float* AB = (float*)d_ws;                 // [384][64]
  float* T  = AB + NLOC * TWO_DHID;         // [384][32][64]

  stage1_linear_in<<<(NLOC * TWO_DHID) / 256, 256, 0, stream>>>(m, W_in, b_in, op_mask, AB);
  stage2_bw<<<(NLOC * D_HID * D_PAIR) / 256, 256, 0, stream>>>(AB, W_out, T);
  stage3_wmma<<<dim3(NLOC, 3), 256, 0, stream>>>(AB, T, b_out, op_norm, z);
}